// RWKV7Adapter_85237920957009
// MI455X (gfx1250) — compile-verified
//
#include <hip/hip_runtime.h>
#include <math.h>

// ---------------- constants (match reference) ----------------
#define D_MODEL 256
#define B_BATCH 4
#define T_SEQ   2048
#define NTOK    (B_BATCH * T_SEQ)          // 8192 tokens
#define DECAY_SCALE (-0.6065306597126334f)
#define GN_EPS  (256.0f * 1e-5f)

typedef __attribute__((ext_vector_type(2))) float v2f;
typedef __attribute__((ext_vector_type(8))) float v8f;

enum { EPI_NONE = 0, EPI_TANH = 1, EPI_SIGMOID = 2, EPI_DECAY = 3 };

// ---------------- token shift / mix ----------------
__global__ __launch_bounds__(256) void rwkv7_mix(
    const float* __restrict__ x, const float* __restrict__ xmix,
    float* __restrict__ xr, float* __restrict__ xw, float* __restrict__ xk,
    float* __restrict__ xv, float* __restrict__ xa, float* __restrict__ xg) {
  int idx = blockIdx.x * 256 + threadIdx.x;     // over NTOK*D
  int d   = idx & (D_MODEL - 1);
  int tok = idx >> 8;
  int t   = tok & (T_SEQ - 1);
  float xc = x[idx];
  float sh = (t > 0) ? x[idx - D_MODEL] : 0.0f;
  float delta = sh - xc;
  xr[idx] = xc + delta * xmix[0 * D_MODEL + d];
  xw[idx] = xc + delta * xmix[1 * D_MODEL + d];
  xk[idx] = xc + delta * xmix[2 * D_MODEL + d];
  xv[idx] = xc + delta * xmix[3 * D_MODEL + d];
  xa[idx] = xc + delta * xmix[4 * D_MODEL + d];
  xg[idx] = xc + delta * xmix[5 * D_MODEL + d];
}

// ---------------- fp32 WMMA GEMM: C = act(A[NTOKxK] @ W[KxN] + bias) ----------
// Templated on N,K,MODE. 8 wave32 per block; block tile 128x16; double-buffered
// LDS staging so next chunk's global->LDS overlaps current chunk's WMMAs.
template<int N, int K, int MODE>
__global__ __launch_bounds__(256) void wmma_gemm_f32(
    const float* __restrict__ A, const float* __restrict__ W,
    const float* __restrict__ bias, float* __restrict__ C) {
  constexpr int KC = 32;                 // K-chunk
  constexpr int AS = 36;                 // padded A-tile row stride (conflict-free)
  constexpr int NC = K / KC;             // number of chunks
  __shared__ float Atile[2][128 * AS];   // 2 x 18 KB
  __shared__ float Wtile[2][KC * 16];    // 2 x 2 KB

  const int tid  = threadIdx.x;
  const int lane = tid & 31;
  const int wave = tid >> 5;
  const int m0   = blockIdx.y * 128;
  const int n0   = blockIdx.x * 16;
  const int mrow = lane & 15;            // A row within the wave's 16x16 tile
  const int ncol = lane & 15;            // B/C column within tile
  const int koff = (lane >> 4) << 1;     // half-wave K split: 0 or 2
  const int m_off = (lane >> 4) * 8;     // C layout: lanes 16-31 hold M=8..15

  auto stage = [&](int k0, int buf) {
#pragma unroll
    for (int i = 0; i < 4; ++i) {
      int f4  = tid + i * 256;
      int row = f4 >> 3;
      int c4  = (f4 & 7) << 2;
      float4 av = *reinterpret_cast<const float4*>(A + (size_t)(m0 + row) * K + k0 + c4);
      *reinterpret_cast<float4*>(&Atile[buf][row * AS + c4]) = av;
    }
    if (tid < 128) {
      int row = tid >> 2;
      int c4  = (tid & 3) << 2;
      float4 wv = *reinterpret_cast<const float4*>(W + (size_t)(k0 + row) * N + n0 + c4);
      *reinterpret_cast<float4*>(&Wtile[buf][row * 16 + c4]) = wv;
    }
  };

  v8f acc = {};
  stage(0, 0);
  __syncthreads();
  for (int c = 0; c < NC; ++c) {
    if (c + 1 < NC) stage((c + 1) * KC, (c + 1) & 1);   // overlap with WMMAs below
    const float* At = &Atile[c & 1][(wave * 16 + mrow) * AS];
    const float* Wt = &Wtile[c & 1][0];
#pragma unroll
    for (int kk0 = 0; kk0 < KC; kk0 += 4) {
      v2f av, bv;
      av.x = At[kk0 + koff];
      av.y = At[kk0 + koff + 1];
      bv.x = Wt[(kk0 + koff) * 16 + ncol];
      bv.y = Wt[(kk0 + koff + 1) * 16 + ncol];
      acc = __builtin_amdgcn_wmma_f32_16x16x4_f32(
          false, av, false, bv, (short)0, acc, false, false);
    }
    __syncthreads();   // readers done with buf c; writers done staging buf c+1
  }

  const float bval = bias ? bias[n0 + ncol] : 0.0f;
#pragma unroll
  for (int i = 0; i < 8; ++i) {
    float val = acc[i] + bval;
    if (MODE == EPI_TANH)         val = tanhf(val);
    else if (MODE == EPI_SIGMOID) val = 1.0f / (1.0f + expf(-val));
    else if (MODE == EPI_DECAY)   val = expf(DECAY_SCALE * (1.0f / (1.0f + expf(-val))));
    C[(size_t)(m0 + m_off + i) * N + n0 + ncol] = val;
  }
}

// ---------------- kk = L2norm(k*k_k), kmod = k*(1+(a-1)*k_a) ----------------
__global__ __launch_bounds__(256) void rwkv7_prep(
    const float* __restrict__ k_in, const float* __restrict__ a,
    const float* __restrict__ k_k, const float* __restrict__ k_a,
    float* __restrict__ kk_out, float* __restrict__ kmod_out) {
  const int tok = blockIdx.x, d = threadIdx.x;
  const size_t idx = (size_t)tok * D_MODEL + d;
  float kv  = k_in[idx];
  float kkv = kv * k_k[d];
  __shared__ float red[256];
  red[d] = kkv * kkv; __syncthreads();
  for (int s = 128; s > 0; s >>= 1) { if (d < s) red[d] += red[d + s]; __syncthreads(); }
  float nrm = fmaxf(sqrtf(red[0]), 1e-12f);
  kk_out[idx]   = kkv / nrm;
  kmod_out[idx] = kv * (1.0f + (a[idx] - 1.0f) * k_a[d]);
}

// ---------------- sequential RWKV7 scan, register-resident state ------------
// 256 V-rows split over 8 WGs/batch (32 blocks). 128 threads/block: 4 threads
// per row, each owns 64 columns in 16 float4 registers. Per-timestep vectors
// double-buffered in registers (load t+1 while computing t) and staged in LDS
// with a 4-float pad every 64 elements for conflict-free quad access.
#define SCAN_WGS_PER_B 8
#define SCAN_ROWS      32
#define SCAN_THREADS   128
#define VPAD(i) ((i) + (((i) >> 6) << 2))      // padded vector index
#define VLEN    (D_MODEL + 12)                 // 268 floats per vector array

__global__ __launch_bounds__(SCAN_THREADS) void rwkv7_scan(
    const float* __restrict__ r,  const float* __restrict__ dec,
    const float* __restrict__ k,  const float* __restrict__ v,
    const float* __restrict__ kk, const float* __restrict__ a,
    float* __restrict__ o) {
  __shared__ float vr[VLEN], vd[VLEN], vk[VLEN], vkk[VLEN], vb[VLEN], vv[VLEN];

  const int tid = threadIdx.x;
  const int b   = blockIdx.x / SCAN_WGS_PER_B;
  const int wg  = blockIdx.x % SCAN_WGS_PER_B;
  const int row_local = tid >> 2;              // 0..31
  const int seg = tid & 3;                     // which 64-column segment
  const int row = wg * SCAN_ROWS + row_local;  // 0..255 (V dimension)
  const int segoff = seg * 68;                 // LDS offset of this segment
  const int rowp = VPAD(row);
  const int p0 = VPAD(tid);                    // this thread's 2 staging slots
  const int p1 = VPAD(tid + SCAN_THREADS);

  float4 s[16];
#pragma unroll
  for (int i = 0; i < 16; ++i) s[i] = make_float4(0.0f, 0.0f, 0.0f, 0.0f);

  const size_t base = (size_t)b * T_SEQ * D_MODEL;

  // preload t=0 vectors into registers
  size_t off0 = base + tid;
  size_t off1 = off0 + SCAN_THREADS;
  float pr0 = r[off0],  pd0 = dec[off0], pk0 = k[off0];
  float pkk0 = kk[off0], pa0 = a[off0],  pv0 = v[off0];
  float pr1 = r[off1],  pd1 = dec[off1], pk1 = k[off1];
  float pkk1 = kk[off1], pa1 = a[off1],  pv1 = v[off1];

  for (int t = 0; t < T_SEQ; ++t) {
    // publish staged registers for step t
    vr[p0] = pr0;  vd[p0] = pd0; vk[p0] = pk0;
    vkk[p0] = pkk0; vb[p0] = pkk0 * pa0; vv[p0] = pv0;
    vr[p1] = pr1;  vd[p1] = pd1; vk[p1] = pk1;
    vkk[p1] = pkk1; vb[p1] = pkk1 * pa1; vv[p1] = pv1;
    __syncthreads();

    // issue loads for step t+1 (overlap with compute below)
    if (t + 1 < T_SEQ) {
      size_t n0 = base + (size_t)(t + 1) * D_MODEL + tid;
      size_t n1 = n0 + SCAN_THREADS;
      pr0 = r[n0];  pd0 = dec[n0]; pk0 = k[n0];
      pkk0 = kk[n0]; pa0 = a[n0];  pv0 = v[n0];
      pr1 = r[n1];  pd1 = dec[n1]; pk1 = k[n1];
      pkk1 = kk[n1]; pa1 = a[n1];  pv1 = v[n1];
    }

    // sa[row] = -dot(S[row,:], kk): partial over my 64 cols, reduce over quad
    float sa = 0.0f;
#pragma unroll
    for (int j = 0; j < 16; ++j) {
      float4 K4 = *reinterpret_cast<const float4*>(vkk + segoff + j * 4);
      sa += s[j].x * K4.x + s[j].y * K4.y + s[j].z * K4.z + s[j].w * K4.w;
    }
    sa += __shfl_xor(sa, 1);
    sa += __shfl_xor(sa, 2);
    sa = -sa;

    // S = S*dec + sa*(kk*a) + v*k ; o = S @ r   (fused, state in registers)
    const float vt = vv[rowp];
    float oacc = 0.0f;
#pragma unroll
    for (int j = 0; j < 16; ++j) {
      const int p = segoff + j * 4;
      float4 d4 = *reinterpret_cast<const float4*>(vd + p);
      float4 b4 = *reinterpret_cast<const float4*>(vb + p);
      float4 k4 = *reinterpret_cast<const float4*>(vk + p);
      float4 r4 = *reinterpret_cast<const float4*>(vr + p);
      s[j].x = s[j].x * d4.x + sa * b4.x + vt * k4.x;
      s[j].y = s[j].y * d4.y + sa * b4.y + vt * k4.y;
      s[j].z = s[j].z * d4.z + sa * b4.z + vt * k4.z;
      s[j].w = s[j].w * d4.w + sa * b4.w + vt * k4.w;
      oacc += s[j].x * r4.x + s[j].y * r4.y + s[j].z * r4.z + s[j].w * r4.w;
    }
    oacc += __shfl_xor(oacc, 1);
    oacc += __shfl_xor(oacc, 2);
    if (seg == 0) o[base + (size_t)t * D_MODEL + row] = oacc;
    __syncthreads();
  }
}

// ---------------- groupnorm + bonus + gate ----------------
__global__ __launch_bounds__(256) void rwkv7_post(
    const float* __restrict__ o, const float* __restrict__ r,
    const float* __restrict__ kmod, const float* __restrict__ v,
    const float* __restrict__ g, const float* __restrict__ r_k,
    const float* __restrict__ gn_w, const float* __restrict__ gn_b,
    float* __restrict__ og) {
  const int tok = blockIdx.x, d = threadIdx.x;
  const size_t idx = (size_t)tok * D_MODEL + d;
  float ov = o[idx];
  __shared__ float red[256];

  red[d] = ov; __syncthreads();
  for (int s = 128; s > 0; s >>= 1) { if (d < s) red[d] += red[d + s]; __syncthreads(); }
  float mu = red[0] * (1.0f / D_MODEL); __syncthreads();

  float c = ov - mu;
  red[d] = c * c; __syncthreads();
  for (int s = 128; s > 0; s >>= 1) { if (d < s) red[d] += red[d + s]; __syncthreads(); }
  float var = red[0] * (1.0f / D_MODEL); __syncthreads();

  float nrm = c * rsqrtf(var + GN_EPS) * gn_w[d] + gn_b[d];

  red[d] = r[idx] * kmod[idx] * r_k[d]; __syncthreads();
  for (int s = 128; s > 0; s >>= 1) { if (d < s) red[d] += red[d + s]; __syncthreads(); }
  float bonus = red[0];

  og[idx] = (nrm + bonus * v[idx]) * g[idx];
}

// ---------------- host-side launch ----------------
template<int N, int K, int MODE>
static inline void launch_gemm(const float* A, const float* W, const float* bias,
                               float* C, hipStream_t stream) {
  dim3 grid((unsigned)(N / 16), (unsigned)(NTOK / 128));
  wmma_gemm_f32<N, K, MODE><<<grid, dim3(256), 0, stream>>>(A, W, bias, C);
}

extern "C" void kernel_launch(void* const* d_in, const int* in_sizes, int n_in,
                              void* d_out, int out_size, void* d_ws, size_t ws_size,
                              hipStream_t stream) {
  (void)in_sizes; (void)n_in; (void)out_size; (void)ws_size;
  const float* x    = (const float*)d_in[0];
  const float* xmix = (const float*)d_in[1];
  const float* W_r  = (const float*)d_in[2];
  const float* W_k  = (const float*)d_in[3];
  const float* W_v  = (const float*)d_in[4];
  const float* W_o  = (const float*)d_in[5];
  const float* A_w  = (const float*)d_in[6];
  const float* B_w  = (const float*)d_in[7];
  const float* b_w  = (const float*)d_in[8];
  const float* A_a  = (const float*)d_in[9];
  const float* B_a  = (const float*)d_in[10];
  const float* b_a  = (const float*)d_in[11];
  const float* A_g  = (const float*)d_in[12];
  const float* B_g  = (const float*)d_in[13];
  const float* k_k  = (const float*)d_in[14];
  const float* k_a  = (const float*)d_in[15];
  const float* r_k  = (const float*)d_in[16];
  const float* gn_w = (const float*)d_in[17];
  const float* gn_b = (const float*)d_in[18];
  float* out = (float*)d_out;
  float* ws  = (float*)d_ws;

  const size_t SZ = (size_t)NTOK * D_MODEL;  // 2,097,152 floats per (B,T,D) buffer
  float* xr    = ws + 0 * SZ;
  float* xw    = ws + 1 * SZ;
  float* xk    = ws + 2 * SZ;
  float* xv    = ws + 3 * SZ;
  float* xa    = ws + 4 * SZ;
  float* xg    = ws + 5 * SZ;
  float* rbuf  = ws + 6 * SZ;
  float* kbuf  = ws + 7 * SZ;
  float* vbuf  = ws + 8 * SZ;
  float* decay = ws + 9 * SZ;
  float* abuf  = ws + 10 * SZ;
  float* gbuf  = ws + 11 * SZ;
  float* kkbuf = ws + 12 * SZ;
  float* kmod  = ws + 13 * SZ;
  float* obuf  = ws + 14 * SZ;
  float* ogbuf = ws + 15 * SZ;
  float* ltmp  = ws + 16 * SZ;               // NTOK x 128 max LoRA intermediate

  dim3 blk(256);

  // 1) token shift + mix
  rwkv7_mix<<<(unsigned)(SZ / 256), blk, 0, stream>>>(x, xmix, xr, xw, xk, xv, xa, xg);

  // 2) main projections (fp32 WMMA, double-buffered LDS staging)
  launch_gemm<256, 256, EPI_NONE>(xr, W_r, nullptr, rbuf, stream);
  launch_gemm<256, 256, EPI_NONE>(xk, W_k, nullptr, kbuf, stream);
  launch_gemm<256, 256, EPI_NONE>(xv, W_v, nullptr, vbuf, stream);

  // 3) LoRA chains with fused epilogues
  launch_gemm<64, 256, EPI_TANH>(xw, A_w, nullptr, ltmp, stream);     // tanh(xw@A_w)
  launch_gemm<256, 64, EPI_DECAY>(ltmp, B_w, b_w, decay, stream);     // exp(s*sigm(.+b_w))
  launch_gemm<64, 256, EPI_NONE>(xa, A_a, nullptr, ltmp, stream);     // xa@A_a
  launch_gemm<256, 64, EPI_SIGMOID>(ltmp, B_a, b_a, abuf, stream);    // sigm(.@B_a+b_a)
  launch_gemm<128, 256, EPI_SIGMOID>(xg, A_g, nullptr, ltmp, stream); // sigm(xg@A_g)
  launch_gemm<256, 128, EPI_NONE>(ltmp, B_g, nullptr, gbuf, stream);  // .@B_g

  // 4) kk normalize + k modulate
  rwkv7_prep<<<NTOK, blk, 0, stream>>>(kbuf, abuf, k_k, k_a, kkbuf, kmod);

  // 5) sequential scan, state in registers, 8 WGs per batch
  rwkv7_scan<<<B_BATCH * SCAN_WGS_PER_B, dim3(SCAN_THREADS), 0, stream>>>(
      rbuf, decay, kmod, vbuf, kkbuf, abuf, obuf);

  // 6) groupnorm + bonus + gate
  rwkv7_post<<<NTOK, blk, 0, stream>>>(obuf, rbuf, kmod, vbuf, gbuf, r_k, gn_w, gn_b, ogbuf);

  // 7) output projection
  launch_gemm<256, 256, EPI_NONE>(ogbuf, W_o, nullptr, out, stream);
}